// MeshDecoder_6012954214525
// MI455X (gfx1250) — compile-verified
//
#include <hip/hip_runtime.h>

typedef __attribute__((ext_vector_type(2))) float v2f;
typedef __attribute__((ext_vector_type(8))) float v8f;

#define NT 128          // edge (N) tile per workgroup
#define MT 64           // out-channel (M) tile per workgroup
#define KC 16           // channels per K chunk
#define KR (KC * 5)     // 80 K-rows per chunk
#define SA_STRIDE (KR + 1)   // pad -> conflict-free A fragment reads
#define SB_STRIDE (NT + 16)  // pad -> half-wave K-rows hit disjoint banks

// Fused gather + GEMM mesh convolution using V_WMMA_F32_16X16X4_F32.
//   out[b, o, e] = act( sum_{c,k} f[b,c,e,k] * W[o, c*5+k] + bias[o] )
// f built on the fly from 5 gathered columns of x (indices: up∘gemm or gemm).
__global__ __launch_bounds__(256) void meshconv_wmma_f32(
    const float* __restrict__ x,     // [B, C, E_in]
    const float* __restrict__ W,     // [O, C*5] row-major
    const float* __restrict__ bias,  // [O]
    const int*   __restrict__ up,    // [..] or nullptr (identity)
    const int*   __restrict__ gemm,  // [E, 4]
    float*       __restrict__ out,   // [B, *, E], channel stride E
    int C, int O, int E, int E_in, int out_bstride, int relu)
{
    __shared__ float sA[MT * SA_STRIDE];
    __shared__ float sB[KR * SB_STRIDE];
    __shared__ int   sIdx[5 * NT];

    const int tid  = threadIdx.x;
    const int lane = tid & 31;
    const int wave = tid >> 5;
    const int wm   = wave >> 2;          // 0..1 -> M offset wm*32
    const int wn   = wave & 3;           // 0..3 -> N offset wn*32
    const int e0   = blockIdx.x * NT;
    const int m0   = blockIdx.y * MT;
    const int b    = blockIdx.z;

    // ---- compose gather indices for this edge tile ----
    for (int t = tid; t < 5 * NT; t += 256) {
        int j  = t / NT;
        int ee = e0 + (t - j * NT);
        int ec = (ee < E) ? ee : (E - 1);          // clamp; masked at store
        int src = (j == 0) ? ec : gemm[ec * 4 + (j - 1)];
        if (up) src = up[src];
        sIdx[t] = src;
    }
    __syncthreads();

    v8f acc[2][2];
    #pragma unroll
    for (int tm = 0; tm < 2; ++tm)
        #pragma unroll
        for (int tn = 0; tn < 2; ++tn)
            #pragma unroll
            for (int r = 0; r < 8; ++r) acc[tm][tn][r] = 0.0f;

    const float* xb = x + (size_t)b * C * E_in;
    const int K = C * 5;
    const int nChunks = K / KR;          // C divisible by 16 for all stages

    const int col     = tid & (NT - 1);  // edge column this thread gathers
    const int halfsel = tid >> 7;        // 0/1 -> which 8 channels of the chunk
    const int ln16    = lane & 15;
    const int kb      = (lane >> 4) << 1;

    const int i0 = sIdx[0 * NT + col];
    const int i1 = sIdx[1 * NT + col];
    const int i2 = sIdx[2 * NT + col];
    const int i3 = sIdx[3 * NT + col];
    const int i4 = sIdx[4 * NT + col];

    for (int ch = 0; ch < nChunks; ++ch) {
        const int cbase = ch * KC;

        // ---- A tile: W[m0..m0+MT) x 80 contiguous K cols ----
        for (int t = tid; t < MT * KR; t += 256) {
            int r = t / KR;
            int k = t - r * KR;
            int m = m0 + r;
            sA[r * SA_STRIDE + k] =
                (m < O) ? W[(size_t)m * K + cbase * 5 + k] : 0.0f;
        }

        // ---- B tile: gather + build 5 features for KC channels x NT edges ----
        #pragma unroll
        for (int q = 0; q < KC / 2; ++q) {
            int cc = halfsel * (KC / 2) + q;
            const float* xc = xb + (size_t)(cbase + cc) * E_in;
            float ve = xc[i0];
            float n0 = xc[i1];
            float n1 = xc[i2];
            float n2 = xc[i3];
            float n3 = xc[i4];
            float* rowp = sB + (cc * 5) * SB_STRIDE + col;
            rowp[0 * SB_STRIDE] = ve;
            rowp[1 * SB_STRIDE] = n0 + n2;
            rowp[2 * SB_STRIDE] = n1 + n3;
            rowp[3 * SB_STRIDE] = fabsf(n0 - n2);
            rowp[4 * SB_STRIDE] = fabsf(n1 - n3);
        }
        __syncthreads();

        // ---- 20 K-steps of 4, 2x2 WMMA tiles per wave ----
        for (int kk = 0; kk < KR; kk += 4) {
            v2f af[2], bf[2];
            #pragma unroll
            for (int tm = 0; tm < 2; ++tm) {
                const float* p =
                    sA + (wm * 32 + tm * 16 + ln16) * SA_STRIDE + kk + kb;
                af[tm][0] = p[0];
                af[tm][1] = p[1];
            }
            #pragma unroll
            for (int tn = 0; tn < 2; ++tn) {
                const float* p =
                    sB + (kk + kb) * SB_STRIDE + wn * 32 + tn * 16 + ln16;
                bf[tn][0] = p[0];
                bf[tn][1] = p[SB_STRIDE];
            }
            #pragma unroll
            for (int tm = 0; tm < 2; ++tm)
                #pragma unroll
                for (int tn = 0; tn < 2; ++tn)
                    acc[tm][tn] = __builtin_amdgcn_wmma_f32_16x16x4_f32(
                        false, af[tm], false, bf[tn],
                        (short)0, acc[tm][tn], false, false);
        }
        __syncthreads();
    }

    // ---- epilogue: bias (+ReLU), bounds-guarded store ----
    #pragma unroll
    for (int tm = 0; tm < 2; ++tm) {
        #pragma unroll
        for (int tn = 0; tn < 2; ++tn) {
            int e = e0 + wn * 32 + tn * 16 + ln16;
            if (e < E) {
                int mbase = m0 + wm * 32 + tm * 16 + ((lane >> 4) << 3);
                #pragma unroll
                for (int r = 0; r < 8; ++r) {
                    int m = mbase + r;
                    if (m < O) {
                        float v = acc[tm][tn][r] + bias[m];
                        if (relu) v = fmaxf(v, 0.0f);
                        out[(size_t)b * out_bstride + (size_t)m * E + e] = v;
                    }
                }
            }
        }
    }
}

// skip -> high half of concat buffer: cat[b][Cout + c][e] = skip[b][c][e]
__global__ void copy_skip_kernel(const float* __restrict__ skip,
                                 float* __restrict__ catb,
                                 int Cout, int E, int total)
{
    int idx = blockIdx.x * blockDim.x + threadIdx.x;
    if (idx >= total) return;
    int per_b = Cout * E;
    int b   = idx / per_b;
    int rem = idx - b * per_b;
    catb[(size_t)b * (2 * per_b) + per_b + rem] = skip[idx];
}

extern "C" void kernel_launch(void* const* d_in, const int* in_sizes, int n_in,
                              void* d_out, int out_size, void* d_ws, size_t ws_size,
                              hipStream_t stream)
{
    (void)in_sizes; (void)n_in; (void)out_size; (void)ws_size;

    const float* fe = (const float*)d_in[0];
    const int CHv[5] = {512, 256, 128, 64, 32};
    const int ESv[4] = {9000, 18000, 36000, 72000};

    // workspace layout (floats): all sizes constant across stages
    float* ws   = (float*)d_ws;
    float* catb = ws;                   // B * cin  * E = 18,432,000
    float* t0   = ws + 18432000;        // B * cout * E =  9,216,000
    float* t1   = t0 + 9216000;         // B * cout * E =  9,216,000

    const float* x = fe;
    int Eprev = 4500;

    for (int i = 0; i < 4; ++i) {
        const float* skip = (const float*)d_in[1 + i * 7 + 0];
        const int*   up   = (const int*)  d_in[1 + i * 7 + 1];
        const int*   gm   = (const int*)  d_in[1 + i * 7 + 2];
        const float* W1   = (const float*)d_in[1 + i * 7 + 3];
        const float* b1   = (const float*)d_in[1 + i * 7 + 4];
        const float* W2   = (const float*)d_in[1 + i * 7 + 5];
        const float* b2   = (const float*)d_in[1 + i * 7 + 6];

        const int cin = CHv[i], cout = CHv[i + 1], E = ESv[i];

        dim3 blk(256, 1, 1);
        dim3 g((E + NT - 1) / NT, (cout + MT - 1) / MT, 4);

        // conv A: x (gathered via up∘gemm) -> cat[:, 0:cout, :]
        meshconv_wmma_f32<<<g, blk, 0, stream>>>(
            x, W1, b1, up, gm, catb, cin, cout, E, Eprev, 2 * cout * E, 0);

        // concat: skip -> cat[:, cout:2*cout, :]   (2*cout == cin)
        int total = 4 * cout * E;
        copy_skip_kernel<<<(total + 255) / 256, 256, 0, stream>>>(
            skip, catb, cout, E, total);

        // conv B: cat -> t0, ReLU  (gather via gemm)
        meshconv_wmma_f32<<<g, blk, 0, stream>>>(
            catb, W1, b1, nullptr, gm, t0, cin, cout, E, E, cout * E, 1);

        // conv C: t0 -> t1 (or d_out on last stage), ReLU
        float* outp = (i < 3) ? t1 : (float*)d_out;
        meshconv_wmma_f32<<<g, blk, 0, stream>>>(
            t0, W2, b2, nullptr, gm, outp, cout, cout, E, E, cout * E, 1);

        x = t1;
        Eprev = E;
    }
}